// MyDynamicEdgeConv_29386166239376
// MI455X (gfx1250) — compile-verified
//
#include <hip/hip_runtime.h>
#include <hip/hip_bf16.h>

typedef __attribute__((ext_vector_type(2))) float v2f;
typedef __attribute__((ext_vector_type(8))) float v8f;
typedef __attribute__((ext_vector_type(4))) unsigned int u32x4;
typedef __attribute__((ext_vector_type(8))) int i32x8;
typedef __attribute__((ext_vector_type(4))) int i32x4;
typedef __attribute__((address_space(3))) float* lds_fptr;

#define NPB    4096        // points per batch
#define NBATCH 4
#define NTOT   (NPB * NBATCH)
#define KNN    16
#define HID    128

#if __has_builtin(__builtin_amdgcn_tensor_load_to_lds)
#define USE_TDM 1
#else
#define USE_TDM 0
#endif

// ---------------------------------------------------------------------------
// sq[n] = sum_c feat[n][c]^2
// ---------------------------------------------------------------------------
__global__ __launch_bounds__(256) void sqnorm_kernel(const float* __restrict__ feat,
                                                     float* __restrict__ sq, int C) {
  int n = blockIdx.x * blockDim.x + threadIdx.x;
  if (n >= NTOT) return;
  const float4* row = (const float4*)(feat + (size_t)n * C);
  float s = 0.f;
  int c4 = C >> 2;
  for (int i = 0; i < c4; ++i) {
    float4 v = row[i];
    s += v.x * v.x + v.y * v.y + v.z * v.z + v.w * v.w;
  }
  sq[n] = s;
}

// ---------------------------------------------------------------------------
// Fused pairwise-distance (f32 WMMA Gram matrix) + top-16 nearest neighbors.
// One workgroup (4 wave32) owns 32 rows of one batch. Column tiles of 64
// points are double-buffered in LDS and prefetched by the Tensor Data Mover
// (TDM pad_enable reproduces the +4 DWORD/row bank-conflict padding), so the
// next tile's DMA overlaps with WMMA + top-k of the current tile.
// d = sq_r + sq_c - 2 * <x_r, x_c>.
// ---------------------------------------------------------------------------
template <int C>
__global__ __launch_bounds__(128) void knn_kernel(const float* __restrict__ feat,
                                                  const float* __restrict__ sq,
                                                  int* __restrict__ knn_idx) {
  constexpr int ROWS = 32, COLS = 64;
  constexpr int LD   = C + 4;      // padded LDS row stride (matches TDM padding)
  constexpr int LDD  = COLS + 5;   // distance-tile padding
  constexpr int NT   = NPB / COLS;
  constexpr unsigned PAD_I = (C == 128) ? 6u : 5u;  // pad interval = C DWORDs (2^(code+1))
  constexpr unsigned PAD_A = 3u;                    // pad amount = 4 DWORDs

  __shared__ float xc[2][COLS][LD];   // double-buffered column tiles
  __shared__ float xr[ROWS][LD];
  __shared__ float dt[ROWS][LDD];
  __shared__ float cdl[ROWS][65];
  __shared__ int   cil[ROWS][65];
  __shared__ float sqr[ROWS];
  __shared__ float sqc[COLS];

  const int tid  = threadIdx.x;
  const int lane = tid & 31;
  const int wave = tid >> 5;
  const int b    = blockIdx.x / (NPB / ROWS);
  const int row0 = (blockIdx.x % (NPB / ROWS)) * ROWS;
  const float* fb  = feat + (size_t)b * NPB * C;
  const float* sqb = sq + (size_t)b * NPB;

#if USE_TDM
  // Issue a TDM load of one 64-row x C-col tile into LDS buffer `buf`.
  auto tdm_issue = [&](int col0, int buf) {
    unsigned long long ga = (unsigned long long)(const void*)(fb + (size_t)col0 * C);
    unsigned lds_off = (unsigned)(unsigned long long)(lds_fptr)&xc[buf][0][0];
    u32x4 g0;
    g0.x = 1u;                                   // count=1, user descriptor
    g0.y = lds_off;                              // lds_addr (bytes)
    g0.z = (unsigned)ga;                         // global_addr[31:0]
    g0.w = (unsigned)((ga >> 32) & 0x01FFFFFFull) | (2u << 30);  // addr[56:32] | type=2
    i32x8 g1;
    g1[0] = (int)((2u << 16) |                   // data_size = 4 bytes
                  (1u << 20) |                   // pad_enable
                  (PAD_I << 22) | (PAD_A << 25));
    g1[1] = (int)(((unsigned)C & 0xFFFFu) << 16);        // tensor_dim0[15:0]
    g1[2] = (int)(((unsigned)NPB & 0xFFFFu) << 16);      // dim0[31:16]=0 | tensor_dim1[15:0]
    g1[3] = (int)(((unsigned)C) << 16);                  // dim1[31:16]=0 | tile_dim0
    g1[4] = COLS;                                        // tile_dim1 | tile_dim2=0
    g1[5] = C;                                           // tensor_dim0_stride[31:0]
    g1[6] = 0;                                           // stride hi | dim1_stride lo
    g1[7] = 0;
    i32x4 z4 = {0, 0, 0, 0};
#if defined(__clang_major__) && __clang_major__ >= 23
    i32x8 z8 = {0, 0, 0, 0, 0, 0, 0, 0};
    __builtin_amdgcn_tensor_load_to_lds(g0, g1, z4, z4, z8, 0);
#else
    __builtin_amdgcn_tensor_load_to_lds(g0, g1, z4, z4, 0);
#endif
  };
#endif

  // Load the 32-row tile (coalesced float4 into padded LDS rows).
  for (int i = tid * 4; i < ROWS * C; i += 128 * 4) {
    int r = i / C, c = i % C;
    *(float4*)&xr[r][c] = *(const float4*)(fb + (size_t)(row0 + r) * C + c);
  }
  if (tid < ROWS) sqr[tid] = sqb[row0 + tid];

#if USE_TDM
  if (wave == 0) tdm_issue(0, 0);   // prefetch tile 0
#endif

  // Per-thread running top-16: thread (row = tid&31, quarter = tid>>5) scans
  // 16 columns of each 64-wide distance tile.
  const int srow = tid & 31;
  const int sqtr = tid >> 5;

  float bd[KNN]; int bi[KNN];
#pragma unroll
  for (int s = 0; s < KNN; ++s) { bd[s] = 3.0e38f; bi[s] = 0; }

  for (int t = 0; t < NT; ++t) {
    const int col0 = t * COLS;
    const int p = t & 1;

#if USE_TDM
    if (t + 1 < NT) {
      if (wave == 0) tdm_issue(col0 + COLS, 1 - p);     // prefetch next tile
      if (tid < COLS) sqc[tid] = sqb[col0 + tid];
      __builtin_amdgcn_s_wait_tensorcnt(1);             // tile t done; t+1 in flight
    } else {
      if (tid < COLS) sqc[tid] = sqb[col0 + tid];
      __builtin_amdgcn_s_wait_tensorcnt(0);
    }
#else
#pragma clang loop unroll(disable)
    for (int i = tid * 4; i < COLS * C; i += 128 * 4) {
      int r = i / C, c = i % C;
      *(float4*)&xc[p][r][c] = *(const float4*)(fb + (size_t)(col0 + r) * C + c);
    }
    if (tid < COLS) sqc[tid] = sqb[col0 + tid];
#endif
    __syncthreads();   // tile + sqc visible to all waves

    // 32x64 distance tile = 2x4 grid of 16x16 WMMA tiles; each wave does 2.
#pragma unroll
    for (int tt = 0; tt < 2; ++tt) {
      const int st   = wave + tt * 4;
      const int mi   = st >> 2;                // 0..1  (M tile)
      const int cj   = st & 3;                 // 0..3  (N tile)
      const int ar   = mi * 16 + (lane & 15);  // A row for this lane
      const int bn   = cj * 16 + (lane & 15);  // B col for this lane
      const int koff = 2 * (lane >> 4);        // K sub-offset per half-wave
      v8f acc = {};
#pragma unroll
      for (int k0 = 0; k0 < C; k0 += 4) {
        v2f av = *(const v2f*)&xr[ar][k0 + koff];
        v2f bv = *(const v2f*)&xc[p][bn][k0 + koff];
        acc = __builtin_amdgcn_wmma_f32_16x16x4_f32(
            /*neg_a=*/false, av, /*neg_b=*/false, bv,
            /*c_mod=*/(short)0, acc, /*reuse_a=*/false, /*reuse_b=*/false);
      }
      const float scn = sqc[bn];
#pragma unroll
      for (int v = 0; v < 8; ++v) {
        const int m = mi * 16 + v + 8 * (lane >> 4);   // D layout: M = v + 8*(lane/16)
        dt[m][bn] = sqr[m] + scn - 2.0f * acc[v];
      }
    }
    __syncthreads();

    // Top-16 update: 4 threads per row, 16 columns each. The insertion sort
    // stays fully unrolled (register-resident lists), but the candidate loop
    // is kept rolled to bound the hot-loop code footprint (I-cache).
    const int cbase = sqtr * 16;
#pragma clang loop unroll(disable)
    for (int c = 0; c < 16; ++c) {
      const float d = dt[srow][cbase + c];
      if (d < bd[KNN - 1]) {
        float id = d; int ii = col0 + cbase + c;
#pragma unroll
        for (int s = 0; s < KNN; ++s) {
          if (id < bd[s]) {
            float td = bd[s]; int ti = bi[s];
            bd[s] = id; bi[s] = ii; id = td; ii = ti;
          }
        }
      }
    }
    __syncthreads();   // scan done: dt free, buffer p free for tile t+2
  }

  // Merge the 4 partial lists per row (64 candidates -> final 16).
#pragma unroll
  for (int s = 0; s < KNN; ++s) {
    cdl[srow][sqtr * KNN + s] = bd[s];
    cil[srow][sqtr * KNN + s] = bi[s];
  }
  __syncthreads();
  if (tid < ROWS) {
    float md[KNN]; int mi2[KNN];
#pragma unroll
    for (int s = 0; s < KNN; ++s) { md[s] = 3.0e38f; mi2[s] = 0; }
#pragma clang loop unroll(disable)
    for (int c = 0; c < 4 * KNN; ++c) {
      float d = cdl[tid][c];
      if (d < md[KNN - 1]) {
        float id = d; int ii = cil[tid][c];
#pragma unroll
        for (int s = 0; s < KNN; ++s) {
          if (id < md[s]) {
            float td = md[s]; int ti = mi2[s];
            md[s] = id; mi2[s] = ii; id = td; ii = ti;
          }
        }
      }
    }
    int* op = knn_idx + ((size_t)b * NPB + row0 + tid) * KNN;
#pragma unroll
    for (int s = 0; s < KNN; ++s) op[s] = b * NPB + mi2[s];  // global node id
  }
}

// ---------------------------------------------------------------------------
// p[n][h] = b[h] + sum_c x[n][c]*(Wi[h][c]-Wj[h][c]);  q[n][h] = sum_c x*Wj.
// W is [HID][2C] row-major; weights stay hot in cache (<=128KB).
// ---------------------------------------------------------------------------
__global__ __launch_bounds__(HID) void linear_pq_kernel(const float* __restrict__ feat,
                                                        const float* __restrict__ W,
                                                        const float* __restrict__ bias,
                                                        float* __restrict__ p,
                                                        float* __restrict__ q, int C) {
  __shared__ float xs[HID];
  const int n = blockIdx.x;
  const int h = threadIdx.x;
  const float* xrow = feat + (size_t)n * C;
  for (int c = h; c < C; c += HID) xs[c] = xrow[c];
  __syncthreads();
  const float* wr = W + (size_t)h * 2 * C;
  float pa = bias[h], qa = 0.f;
  for (int c = 0; c < C; ++c) {
    const float xv = xs[c];
    const float wi = wr[c];
    const float wj = wr[C + c];
    pa = fmaf(xv, wi - wj, pa);
    qa = fmaf(xv, wj, qa);
  }
  p[(size_t)n * HID + h] = pa;
  q[(size_t)n * HID + h] = qa;
}

// ---------------------------------------------------------------------------
// out[n][h] = relu?( p[n][h] + max_j q[idx[n][j]][h] )   (p constant over j)
// ---------------------------------------------------------------------------
__global__ __launch_bounds__(HID) void edge_max_kernel(const float* __restrict__ p,
                                                       const float* __restrict__ q,
                                                       const int* __restrict__ knn_idx,
                                                       float* __restrict__ out, int relu) {
  __shared__ int nb[KNN];
  const int n = blockIdx.x;
  const int h = threadIdx.x;
  if (h < KNN) nb[h] = knn_idx[(size_t)n * KNN + h];
  __syncthreads();
  float m = -3.0e38f;
#pragma unroll
  for (int j = 0; j < KNN; ++j) {
    m = fmaxf(m, q[(size_t)nb[j] * HID + h]);
  }
  float r = p[(size_t)n * HID + h] + m;
  if (relu) r = fmaxf(r, 0.f);
  out[(size_t)n * HID + h] = r;
}

// ---------------------------------------------------------------------------
// Driver: 3 EdgeConv layers (C=64 -> 128 -> 128 -> 128)
// ---------------------------------------------------------------------------
extern "C" void kernel_launch(void* const* d_in, const int* in_sizes, int n_in,
                              void* d_out, int out_size, void* d_ws, size_t ws_size,
                              hipStream_t stream) {
  const float* x  = (const float*)d_in[0];
  // d_in[1] = batch assignment (int64), implied by the fixed [B,N] layout.
  const float* W1 = (const float*)d_in[2];
  const float* b1 = (const float*)d_in[3];
  const float* W2 = (const float*)d_in[4];
  const float* b2 = (const float*)d_in[5];
  const float* W3 = (const float*)d_in[6];
  const float* b3 = (const float*)d_in[7];
  float* out = (float*)d_out;

  char* w = (char*)d_ws;
  float* sq  = (float*)w;  w += (size_t)NTOT * sizeof(float);
  int*   idx = (int*)w;    w += (size_t)NTOT * KNN * sizeof(int);
  float* p   = (float*)w;  w += (size_t)NTOT * HID * sizeof(float);
  float* q   = (float*)w;  w += (size_t)NTOT * HID * sizeof(float);
  float* h1  = (float*)w;  w += (size_t)NTOT * HID * sizeof(float);
  float* h2  = (float*)w;  w += (size_t)NTOT * HID * sizeof(float);

  const int knn_grid = NBATCH * (NPB / 32);

  // ---- Layer 1 (C = 64) ----
  sqnorm_kernel<<<NTOT / 256, 256, 0, stream>>>(x, sq, 64);
  knn_kernel<64><<<knn_grid, 128, 0, stream>>>(x, sq, idx);
  linear_pq_kernel<<<NTOT, HID, 0, stream>>>(x, W1, b1, p, q, 64);
  edge_max_kernel<<<NTOT, HID, 0, stream>>>(p, q, idx, h1, 1);

  // ---- Layer 2 (C = 128) ----
  sqnorm_kernel<<<NTOT / 256, 256, 0, stream>>>(h1, sq, 128);
  knn_kernel<128><<<knn_grid, 128, 0, stream>>>(h1, sq, idx);
  linear_pq_kernel<<<NTOT, HID, 0, stream>>>(h1, W2, b2, p, q, 128);
  edge_max_kernel<<<NTOT, HID, 0, stream>>>(p, q, idx, h2, 1);

  // ---- Layer 3 (C = 128, no relu) ----
  sqnorm_kernel<<<NTOT / 256, 256, 0, stream>>>(h2, sq, 128);
  knn_kernel<128><<<knn_grid, 128, 0, stream>>>(h2, sq, idx);
  linear_pq_kernel<<<NTOT, HID, 0, stream>>>(h2, W3, b3, p, q, 128);
  edge_max_kernel<<<NTOT, HID, 0, stream>>>(p, q, idx, out, 0);
}